// GraphSageLinkPred_56624848830738
// MI455X (gfx1250) — compile-verified
//
#include <hip/hip_runtime.h>

namespace {
constexpr int Hdim  = 128;
constexpr int NUSER = 100000;
constexpr int NPROD = 30000;
constexpr int NEDGE = 1000000;
constexpr int NLAB  = 500000;
}

typedef __attribute__((ext_vector_type(2))) float v2f;
typedef __attribute__((ext_vector_type(8))) float v8f;

__device__ __forceinline__ v8f wmma4(v2f a, v2f b, v8f c) {
  // V_WMMA_F32_16X16X4_F32 : exact fp32 multiplies, f32 accumulate
  return __builtin_amdgcn_wmma_f32_16x16x4_f32(false, a, false, b, (short)0, c,
                                               false, false);
}

// ---------------------------------------------------------------------------
// out[M,128] = X[M,K] @ W[K,128] + bias[128] + emb[node_id[m], :]
// One wave = one 16x16 tile; 8 waves/block = 16x128 strip. M % 16 == 0.
// ---------------------------------------------------------------------------
__global__ void k_input_fuse(const float* __restrict__ X,
                             const float* __restrict__ W,
                             const float* __restrict__ bias,
                             const float* __restrict__ emb,
                             const int* __restrict__ nid,
                             float* __restrict__ out, int K) {
  const int lane = threadIdx.x & 31;
  const int wv   = threadIdx.x >> 5;      // column tile 0..7
  const int m0   = blockIdx.x << 4;
  const int n0   = wv << 4;
  const int half = lane >> 4;             // 0: K+0/1, 1: K+2/3
  const int l16  = lane & 15;
  const int n    = n0 + l16;

  const float* Ar = X + (size_t)(m0 + l16) * K;
  v8f c = {};
  for (int kk = 0; kk < K; kk += 4) {
    const int ka = kk + half * 2;
    v2f a, b;
    a.x = Ar[ka];
    a.y = Ar[ka + 1];
    b.x = W[(size_t)ka * Hdim + n];
    b.y = W[(size_t)(ka + 1) * Hdim + n];
    c = wmma4(a, b, c);
  }
  const float bn = bias[n];
#pragma unroll
  for (int r = 0; r < 8; ++r) {
    const int row = m0 + r + half * 8;
    out[(size_t)row * Hdim + n] = c[r] + bn + emb[(size_t)nid[row] * Hdim + n];
  }
}

// ---------------------------------------------------------------------------
// out = relu?( (Agg * invc[row]) @ Wl + bl + Xs @ Wr )   (K = 128)
// Mean-normalization folded into the A-fragment load.
// ---------------------------------------------------------------------------
__global__ void k_layer(const float* __restrict__ Agg,
                        const float* __restrict__ invc,
                        const float* __restrict__ Xs,
                        const float* __restrict__ Wl,
                        const float* __restrict__ bl,
                        const float* __restrict__ Wr,
                        float* __restrict__ out, int relu) {
  const int lane = threadIdx.x & 31;
  const int wv   = threadIdx.x >> 5;
  const int m0   = blockIdx.x << 4;
  const int n0   = wv << 4;
  const int half = lane >> 4;
  const int l16  = lane & 15;
  const int n    = n0 + l16;
  const int arow = m0 + l16;

  const float sc  = invc[arow];
  const float* Ag = Agg + (size_t)arow * Hdim;
  const float* Xr = Xs + (size_t)arow * Hdim;
  v8f c = {};
#pragma unroll 4
  for (int kk = 0; kk < Hdim; kk += 4) {
    const int ka = kk + half * 2;
    v2f a1, b1, a2, b2;
    a1.x = Ag[ka] * sc;
    a1.y = Ag[ka + 1] * sc;
    b1.x = Wl[(size_t)ka * Hdim + n];
    b1.y = Wl[(size_t)(ka + 1) * Hdim + n];
    c = wmma4(a1, b1, c);
    a2.x = Xr[ka];
    a2.y = Xr[ka + 1];
    b2.x = Wr[(size_t)ka * Hdim + n];
    b2.y = Wr[(size_t)(ka + 1) * Hdim + n];
    c = wmma4(a2, b2, c);
  }
  const float bn = bl[n];
#pragma unroll
  for (int r = 0; r < 8; ++r) {
    const int row = m0 + r + half * 8;
    float v = c[r] + bn;
    if (relu) v = fmaxf(v, 0.0f);
    out[(size_t)row * Hdim + n] = v;
  }
}

// ---------------------------------------------------------------------------
// Degree counts (as float, so unsafeAtomicAdd -> global_atomic_add_f32)
// ---------------------------------------------------------------------------
__global__ void k_count(const int* __restrict__ src, const int* __restrict__ dst,
                        float* __restrict__ cu, float* __restrict__ cp, int E) {
  const int e = blockIdx.x * blockDim.x + threadIdx.x;
  if (e >= E) return;
  unsafeAtomicAdd(&cu[src[e]], 1.0f);
  unsafeAtomicAdd(&cp[dst[e]], 1.0f);
}

__global__ void k_inv(float* __restrict__ c, int n) {
  const int i = blockIdx.x * blockDim.x + threadIdx.x;
  if (i < n) c[i] = 1.0f / fmaxf(c[i], 1.0f);
}

// ---------------------------------------------------------------------------
// Scatter-add: one wave per edge; lane = one float4 slice of the 128-f row.
// 512B fully-coalesced gather + 4 fp32 atomics per lane.
// ---------------------------------------------------------------------------
__global__ void k_scatter(const float* __restrict__ feat,
                          const int* __restrict__ gidx,
                          const int* __restrict__ sidx,
                          float* __restrict__ accum, int E) {
  const int e = blockIdx.x * (blockDim.x >> 5) + (threadIdx.x >> 5);
  if (e >= E) return;
  const int lane = threadIdx.x & 31;
  const float4 v =
      *(const float4*)(feat + (size_t)gidx[e] * Hdim + lane * 4);
  float* d = accum + (size_t)sidx[e] * Hdim + lane * 4;
  unsafeAtomicAdd(d + 0, v.x);
  unsafeAtomicAdd(d + 1, v.y);
  unsafeAtomicAdd(d + 2, v.z);
  unsafeAtomicAdd(d + 3, v.w);
}

// ---------------------------------------------------------------------------
// pred[p] = dot(xu[ls[p]], xp[ld[p]]) ; one wave per pair, shfl_xor reduce.
// ---------------------------------------------------------------------------
__global__ void k_dot(const float* __restrict__ xu, const float* __restrict__ xp,
                      const int* __restrict__ ls, const int* __restrict__ ld,
                      float* __restrict__ out, int EL) {
  const int p = blockIdx.x * (blockDim.x >> 5) + (threadIdx.x >> 5);
  if (p >= EL) return;
  const int lane = threadIdx.x & 31;
  const float4 a = *(const float4*)(xu + (size_t)ls[p] * Hdim + lane * 4);
  const float4 b = *(const float4*)(xp + (size_t)ld[p] * Hdim + lane * 4);
  float s = a.x * b.x + a.y * b.y + a.z * b.z + a.w * b.w;
#pragma unroll
  for (int off = 16; off > 0; off >>= 1) s += __shfl_xor(s, off, 32);
  if (lane == 0) out[p] = s;
}

// ---------------------------------------------------------------------------
extern "C" void kernel_launch(void* const* d_in, const int* in_sizes, int n_in,
                              void* d_out, int out_size, void* d_ws,
                              size_t ws_size, hipStream_t stream) {
  const float* user_x     = (const float*)d_in[0];
  const float* prod_x     = (const float*)d_in[1];
  const float* user_emb   = (const float*)d_in[2];
  const float* prod_emb   = (const float*)d_in[3];
  const float* user_lin_w = (const float*)d_in[4];
  const float* user_lin_b = (const float*)d_in[5];
  const float* prod_lin_w = (const float*)d_in[6];
  const float* prod_lin_b = (const float*)d_in[7];
  const float* Wl         = (const float*)d_in[8];
  const float* bl         = (const float*)d_in[9];
  const float* Wr         = (const float*)d_in[10];
  const int*   user_nid   = (const int*)d_in[11];
  const int*   prod_nid   = (const int*)d_in[12];
  const int*   edge_src   = (const int*)d_in[13];
  const int*   edge_dst   = (const int*)d_in[14];
  const int*   label_src  = (const int*)d_in[15];
  const int*   label_dst  = (const int*)d_in[16];
  float* pred = (float*)d_out;

  // --- workspace carve-out (256B aligned) ---
  size_t off = 0;
  auto carve = [&](size_t bytes) {
    void* p = (char*)d_ws + off;
    off += (bytes + 255) & ~(size_t)255;
    return p;
  };
  const size_t ub = (size_t)NUSER * Hdim * sizeof(float);
  const size_t pb = (size_t)NPROD * Hdim * sizeof(float);
  float* xu_a  = (float*)carve(ub);
  float* xu_b  = (float*)carve(ub);
  float* xp_a  = (float*)carve(pb);
  float* xp_b  = (float*)carve(pb);
  float* agg_u = (float*)carve(ub);
  float* agg_p = (float*)carve(pb);
  float* cnt_u = (float*)carve((size_t)NUSER * sizeof(float));
  float* cnt_p = (float*)carve((size_t)NPROD * sizeof(float));

  // --- degree counts -> inverse mean factors (static per launch) ---
  hipMemsetAsync(cnt_u, 0, (size_t)NUSER * sizeof(float), stream);
  hipMemsetAsync(cnt_p, 0, (size_t)NPROD * sizeof(float), stream);
  k_count<<<(NEDGE + 255) / 256, 256, 0, stream>>>(edge_src, edge_dst, cnt_u,
                                                   cnt_p, NEDGE);
  k_inv<<<(NUSER + 255) / 256, 256, 0, stream>>>(cnt_u, NUSER);
  k_inv<<<(NPROD + 255) / 256, 256, 0, stream>>>(cnt_p, NPROD);

  // --- input fusion (WMMA f32) ---
  k_input_fuse<<<NUSER / 16, 256, 0, stream>>>(user_x, user_lin_w, user_lin_b,
                                               user_emb, user_nid, xu_a, 32);
  k_input_fuse<<<NPROD / 16, 256, 0, stream>>>(prod_x, prod_lin_w, prod_lin_b,
                                               prod_emb, prod_nid, xp_a, 64);

  float* xu_cur = xu_a; float* xu_nxt = xu_b;
  float* xp_cur = xp_a; float* xp_nxt = xp_b;
  const int scat_blocks = (NEDGE + 7) / 8;

  for (int i = 0; i < 3; ++i) {
    const float* Wl0 = Wl + (size_t)(i * 2 + 0) * Hdim * Hdim;
    const float* Wl1 = Wl + (size_t)(i * 2 + 1) * Hdim * Hdim;
    const float* Wr0 = Wr + (size_t)(i * 2 + 0) * Hdim * Hdim;
    const float* Wr1 = Wr + (size_t)(i * 2 + 1) * Hdim * Hdim;
    const float* bl0 = bl + (size_t)(i * 2 + 0) * Hdim;
    const float* bl1 = bl + (size_t)(i * 2 + 1) * Hdim;
    const int relu = (i < 2) ? 1 : 0;

    // edge type 0: user -> prod
    hipMemsetAsync(agg_p, 0, pb, stream);
    k_scatter<<<scat_blocks, 256, 0, stream>>>(xu_cur, edge_src, edge_dst,
                                               agg_p, NEDGE);
    k_layer<<<NPROD / 16, 256, 0, stream>>>(agg_p, cnt_p, xp_cur, Wl0, bl0,
                                            Wr0, xp_nxt, relu);

    // edge type 1: prod -> user (uses OLD xp, still in xp_cur)
    hipMemsetAsync(agg_u, 0, ub, stream);
    k_scatter<<<scat_blocks, 256, 0, stream>>>(xp_cur, edge_dst, edge_src,
                                               agg_u, NEDGE);
    k_layer<<<NUSER / 16, 256, 0, stream>>>(agg_u, cnt_u, xu_cur, Wl1, bl1,
                                            Wr1, xu_nxt, relu);

    float* t;
    t = xu_cur; xu_cur = xu_nxt; xu_nxt = t;
    t = xp_cur; xp_cur = xp_nxt; xp_nxt = t;
  }

  k_dot<<<(NLAB + 7) / 8, 256, 0, stream>>>(xu_cur, xp_cur, label_src,
                                            label_dst, pred, NLAB);
  (void)in_sizes; (void)n_in; (void)out_size; (void)ws_size;
}